// simple_48378511622250
// MI455X (gfx1250) — compile-verified
//
#include <hip/hip_runtime.h>

#define N_NODES 40000
#define N_EDGES 640000
#define NFEAT   256
#define NHID    128

typedef float v2f __attribute__((ext_vector_type(2)));
typedef float v8f __attribute__((ext_vector_type(8)));

// ---------------------------------------------------------------------------
// K1: support = x @ W using V_WMMA_F32_16X16X4_F32 (fp32 WMMA, wave32).
// Block = 256 threads = 8 waves. Block owns a 16-row strip; wave w owns
// columns [16w, 16w+16). K-loop steps by 4 (the WMMA K), 64 WMMAs per tile.
// A-frag layout (ISA 7.12.2, 32-bit A 16x4): lanes 0-15 -> M=lane, VGPR0/1 =
// K{0,1}; lanes 16-31 -> K{2,3}.  B-frag mirrored: lanes hold N, halves hold
// K{0,1}/{2,3}.  D: VGPR v -> M = m0 + 8*half + v, N = n0 + (lane&15).
// ---------------------------------------------------------------------------
__global__ __launch_bounds__(256) void gcn_gemm_wmma(
    const float* __restrict__ x, const float* __restrict__ W,
    float* __restrict__ support) {
  const int lane = threadIdx.x & 31;
  const int wave = threadIdx.x >> 5;
  const int m0   = blockIdx.x * 16;
  const int n0   = wave * 16;
  const int half = lane >> 4;      // 0 for lanes 0-15, 1 for lanes 16-31
  const int l15  = lane & 15;
  const int koff = half * 2;       // this half's K pair within each K=4 step
  const int row  = m0 + l15;       // A: lane index is M
  const int col  = n0 + l15;       // B: lane index is N

  const float* xr = x + (size_t)row * NFEAT + koff;   // contiguous K pair
  const float* wc = W + (size_t)koff * NHID + col;    // strided K pair

  v8f acc = {};
#pragma unroll 8
  for (int k = 0; k < NFEAT; k += 4) {
    const float2 av = *(const float2*)(xr + k);       // global_load_b64
    v2f a; a.x = av.x; a.y = av.y;
    v2f b; b.x = wc[(size_t)k * NHID];
           b.y = wc[(size_t)(k + 1) * NHID];
    // 8-arg form: (neg_a, A, neg_b, B, c_mod, C, reuse_a, reuse_b)
    acc = __builtin_amdgcn_wmma_f32_16x16x4_f32(
        false, a, false, b, (short)0, acc, false, false);
  }

  float* outp = support + (size_t)(m0 + half * 8) * NHID + col;
#pragma unroll
  for (int v = 0; v < 8; ++v) outp[(size_t)v * NHID] = acc[v];
}

// ---------------------------------------------------------------------------
// K0: out[n][c] = b[c]  (bias folded into the accumulator init, float4-wide)
// ---------------------------------------------------------------------------
__global__ __launch_bounds__(256) void gcn_init_bias(
    float4* __restrict__ out4, const float4* __restrict__ b4) {
  const int i = blockIdx.x * 256 + threadIdx.x;   // over N_NODES * 32 quads
  out4[i] = b4[i & (NHID / 4 - 1)];
}

// ---------------------------------------------------------------------------
// K2: per edge e: out[dst] += support[src] * w   (one wave per edge,
// 32 lanes x float4 = 128 feats; 512B coalesced gather, native f32 atomics
// into L2-resident output).
// ---------------------------------------------------------------------------
__global__ __launch_bounds__(256) void gcn_scatter(
    const float* __restrict__ support, const long long* __restrict__ ei,
    const float* __restrict__ ew, float* __restrict__ out) {
  const int tid  = blockIdx.x * 256 + threadIdx.x;
  const int edge = tid >> 5;
  const int lane = tid & 31;

  const long long src = ei[edge];            // row 0 of edge_index
  const long long dst = ei[N_EDGES + edge];  // row 1 of edge_index
  const float     w   = ew[edge];

  const float4 v = ((const float4*)(support + (size_t)src * NHID))[lane];
  float* o = out + (size_t)dst * NHID + lane * 4;
  unsafeAtomicAdd(o + 0, v.x * w);   // global_atomic_add_f32
  unsafeAtomicAdd(o + 1, v.y * w);
  unsafeAtomicAdd(o + 2, v.z * w);
  unsafeAtomicAdd(o + 3, v.w * w);
}

// ---------------------------------------------------------------------------
// K3: in-place ReLU, float4-wide.
// ---------------------------------------------------------------------------
__global__ __launch_bounds__(256) void gcn_relu(float4* __restrict__ out4) {
  const int i = blockIdx.x * 256 + threadIdx.x;
  float4 v = out4[i];
  v.x = fmaxf(v.x, 0.0f); v.y = fmaxf(v.y, 0.0f);
  v.z = fmaxf(v.z, 0.0f); v.w = fmaxf(v.w, 0.0f);
  out4[i] = v;
}

extern "C" void kernel_launch(void* const* d_in, const int* in_sizes, int n_in,
                              void* d_out, int out_size, void* d_ws,
                              size_t ws_size, hipStream_t stream) {
  const float*     x  = (const float*)d_in[0];      // [40000, 256] f32
  const long long* ei = (const long long*)d_in[1];  // [2, 640000] i64
  const float*     ew = (const float*)d_in[2];      // [640000] f32
  const float*     W  = (const float*)d_in[3];      // [256, 128] f32
  const float*     b  = (const float*)d_in[4];      // [128] f32
  float* out     = (float*)d_out;                   // [40000, 128] f32
  float* support = (float*)d_ws;                    // 20.5 MB scratch

  // 1) support = x @ W  (WMMA fp32)
  gcn_gemm_wmma<<<N_NODES / 16, 256, 0, stream>>>(x, W, support);
  // 2) out = bias (accumulator init)
  gcn_init_bias<<<(N_NODES * NHID / 4) / 256, 256, 0, stream>>>(
      (float4*)out, (const float4*)b);
  // 3) out[dst] += support[src] * w  (atomic scatter-add)
  gcn_scatter<<<(N_EDGES * 32) / 256, 256, 0, stream>>>(support, ei, ew, out);
  // 4) out = relu(out)
  gcn_relu<<<(N_NODES * NHID / 4) / 256, 256, 0, stream>>>((float4*)out);
}